// HausdorffDTLoss_47158740910163
// MI455X (gfx1250) — compile-verified
//
#include <hip/hip_runtime.h>
#include <hip/hip_bf16.h>

#define B_ 4
#define C_ 12
#define H_ 256
#define W_ 256
#define HW_ (H_ * W_)
#define NPIX (B_ * HW_)      // 262144
#define NIMG (2 * B_ * C_)   // 96 masks (48 pred + 48 target)
#define NTASK (2 * NIMG)     // 192 EDTs (each mask and its complement)
#define LARGE_F 1.0e6f
#define LARGE2_F 1.0e12f

typedef float v2f __attribute__((ext_vector_type(2)));
typedef float v8f __attribute__((ext_vector_type(8)));

// workspace layout (bytes)
#define OFF_ACC   0ull                     // 2 doubles
#define OFF_FLAGS 64ull                    // 96 ints (has_fg per mask image)
#define OFF_MASK  1024ull                  // 96 * 65536 bytes
#define OFF_G2    6295552ull               // 192 * 65536 floats (g2 -> then d in-place)
// total ws needed: 6295552 + 50331648 = 56,627,200 bytes

// ---------------------------------------------------------------- init
__global__ void hd_init(double* __restrict__ acc, int* __restrict__ flags) {
  int t = threadIdx.x;
  if (t < 2) acc[t] = 0.0;
  if (t < NIMG) flags[t] = 0;
}

// ------------------------------------------------- masks + has_fg flags
__global__ void hd_masks(const float* __restrict__ pred, const int* __restrict__ tgt,
                         unsigned char* __restrict__ mask, int* __restrict__ flags) {
  int p = blockIdx.x * blockDim.x + threadIdx.x;   // [0, NPIX)
  int b = p >> 16;            // HW_ == 65536
  int hw = p & (HW_ - 1);
  float x[C_];
  float mx = -3.4e38f;
#pragma unroll
  for (int c = 0; c < C_; ++c) {
    x[c] = pred[(size_t)(b * C_ + c) * HW_ + hw];
    mx = fmaxf(mx, x[c]);
  }
  float s = 0.f;
#pragma unroll
  for (int c = 0; c < C_; ++c) { x[c] = expf(x[c] - mx); s += x[c]; }
  int t = tgt[p];
  bool lane0 = ((threadIdx.x & 31) == 0);
#pragma unroll
  for (int c = 0; c < C_; ++c) {
    unsigned char mp = (x[c] / s > 0.5f) ? 1 : 0;
    unsigned char mt = (t == c) ? 1 : 0;
    int ip = b * C_ + c;
    int it = (NIMG / 2) + ip;
    mask[(size_t)ip * HW_ + hw] = mp;
    mask[(size_t)it * HW_ + hw] = mt;
    unsigned long long bp = __ballot((int)mp);
    unsigned long long bt = __ballot((int)mt);
    if (lane0 && bp) atomicOr(&flags[ip], 1);
    if (lane0 && bt) atomicOr(&flags[it], 1);
  }
}

// ---------------------------------- 1D EDT along H (fwd+bwd), write g^2
__global__ void hd_edt_v(const unsigned char* __restrict__ mask, float* __restrict__ g2) {
  int task = blockIdx.x;      // [0, NTASK)
  int img = task >> 1;
  int side = task & 1;        // 0: feature = !mask ; 1: feature = mask
  int w = threadIdx.x;
  const unsigned char* m = mask + (size_t)img * HW_;
  float* g = g2 + (size_t)task * HW_;
  float c = LARGE_F;
  for (int h = 0; h < H_; ++h) {
    if (h + 8 < H_) __builtin_prefetch(m + (h + 8) * W_ + w, 0, 0);  // global_prefetch_b8
    bool fi = ((m[h * W_ + w] != 0) == (side != 0));
    c = fi ? 0.f : fminf(c + 1.f, LARGE_F);
    g[h * W_ + w] = c;                      // fwd distance (temp)
  }
  float cb = LARGE_F;
  for (int h = H_ - 1; h >= 0; --h) {
    bool fi = ((m[h * W_ + w] != 0) == (side != 0));
    cb = fi ? 0.f : fminf(cb + 1.f, LARGE_F);
    float gm = fminf(cb, g[h * W_ + w]);
    g[h * W_ + w] = gm * gm;                // g^2
  }
}

// --------- exact lower envelope along W: d = sqrt(min_j' g2[j'] + (j-j')^2)
__global__ void hd_edt_h(float* __restrict__ g2) {
  __shared__ float row[W_];
  int j = threadIdx.x;
  float* base = g2 + (size_t)blockIdx.x * W_;   // one row (task*256 + h)
  row[j] = base[j];
  __syncthreads();
  float best = LARGE2_F;
#pragma unroll 8
  for (int jp = 0; jp < W_; ++jp) {
    float dj = (float)(j - jp);
    best = fminf(best, fmaf(dj, dj, row[jp]));  // dj*dj exact -> matches mul+add
  }
  base[j] = sqrtf(best);                        // in-place: d (not d^2)
}

// ----------------- fused softmax + CE + loss, WMMA-assisted reduction
__global__ void hd_loss(const float* __restrict__ pred, const int* __restrict__ tgt,
                        const float* __restrict__ dbuf, const int* __restrict__ flags,
                        double* __restrict__ acc) {
  int p = blockIdx.x * blockDim.x + threadIdx.x;   // exactly NPIX threads, no divergence
  int b = p >> 16;
  int hw = p & (HW_ - 1);
  float sh[C_], ex[C_];
  float mx = -3.4e38f;
#pragma unroll
  for (int c = 0; c < C_; ++c) {
    sh[c] = pred[(size_t)(b * C_ + c) * HW_ + hw];
    mx = fmaxf(mx, sh[c]);
  }
  float s = 0.f;
#pragma unroll
  for (int c = 0; c < C_; ++c) { sh[c] -= mx; ex[c] = expf(sh[c]); s += ex[c]; }
  int t = tgt[p];
  float logp_t = sh[t] - logf(s);                  // log_softmax at target

  float accL = 0.f;
#pragma unroll
  for (int c = 0; c < C_; ++c) {
    float pc = ex[c] / s;
    float e = pc - ((t == c) ? 1.f : 0.f);
    int ip = b * C_ + c;
    int it = (NIMG / 2) + ip;
    size_t op = (size_t)(2 * ip) * HW_ + hw;
    size_t ot = (size_t)(2 * it) * HW_ + hw;
    float fp = dbuf[op] + dbuf[op + HW_];          // edt(mask) + edt(~mask)
    float ft = dbuf[ot] + dbuf[ot + HW_];
    fp = flags[ip] ? fp : 0.f;
    ft = flags[it] ? ft : 0.f;
    float dist = fp * fp + ft * ft;                // field^ALPHA, ALPHA=2
    accL += e * e * dist;
  }

  // Wave reduction: one v_wmma_f32_16x16x4_f32 with A=ones does the 32->16
  // pairwise-sum step (D[m][n] = x_n + x_{n+16}, replicated to every lane as
  // s[lane%16]); 4 xor-shuffles finish the wave sum. EXEC is all ones here.
  v2f ones; ones.x = 1.f; ones.y = 1.f;
  v8f cz = {};
  v2f bL; bL.x = accL; bL.y = 0.f;
  v8f dL = __builtin_amdgcn_wmma_f32_16x16x4_f32(false, ones, false, bL,
                                                 (short)0, cz, false, false);
  float sL = dL[0];
  sL += __shfl_xor(sL, 1);
  sL += __shfl_xor(sL, 2);
  sL += __shfl_xor(sL, 4);
  sL += __shfl_xor(sL, 8);

  v2f bC; bC.x = logp_t; bC.y = 0.f;
  v8f dC = __builtin_amdgcn_wmma_f32_16x16x4_f32(false, ones, false, bC,
                                                 (short)0, cz, false, false);
  float sC = dC[0];
  sC += __shfl_xor(sC, 1);
  sC += __shfl_xor(sC, 2);
  sC += __shfl_xor(sC, 4);
  sC += __shfl_xor(sC, 8);

  __shared__ double wl[8], wc[8];
  int wid = threadIdx.x >> 5;
  if ((threadIdx.x & 31) == 0) { wl[wid] = (double)sL; wc[wid] = (double)sC; }
  __syncthreads();
  if (threadIdx.x == 0) {
    double tl = 0.0, tc = 0.0;
#pragma unroll
    for (int i = 0; i < 8; ++i) { tl += wl[i]; tc += wc[i]; }
    atomicAdd(&acc[0], tl);   // global_atomic_add_f64
    atomicAdd(&acc[1], tc);
  }
}

// ---------------------------------------------------------------- finalize
__global__ void hd_final(const double* __restrict__ acc, float* __restrict__ out) {
  if (threadIdx.x == 0) {
    double n = (double)NPIX;
    double loss = acc[0] / n / (double)(C_ * B_ * 3);
    double ce = -acc[1] / n;
    out[0] = (float)(loss + ce);
  }
}

extern "C" void kernel_launch(void* const* d_in, const int* in_sizes, int n_in,
                              void* d_out, int out_size, void* d_ws, size_t ws_size,
                              hipStream_t stream) {
  const float* pred = (const float*)d_in[0];
  const int* target = (const int*)d_in[1];
  float* out = (float*)d_out;
  char* ws = (char*)d_ws;
  double* accum = (double*)(ws + OFF_ACC);
  int* flags = (int*)(ws + OFF_FLAGS);
  unsigned char* mask = (unsigned char*)(ws + OFF_MASK);
  float* g2 = (float*)(ws + OFF_G2);

  hd_init<<<1, 128, 0, stream>>>(accum, flags);
  hd_masks<<<NPIX / 256, 256, 0, stream>>>(pred, target, mask, flags);
  hd_edt_v<<<NTASK, W_, 0, stream>>>(mask, g2);
  hd_edt_h<<<NTASK * H_, W_, 0, stream>>>(g2);
  hd_loss<<<NPIX / 256, 256, 0, stream>>>(pred, target, g2, flags, accum);
  hd_final<<<1, 64, 0, stream>>>(accum, out);
}